// GAGCN_54511724921067
// MI455X (gfx1250) — compile-verified
//
#include <hip/hip_runtime.h>
#include <hip/hip_bf16.h>
#include <math.h>

// ---------------- problem constants (from reference) ----------------
#define BB      2
#define NP      8192          // points per batch
#define KNBR    20            // kept neighbors
#define RATE    2
#define CAND    (KNBR*RATE)   // 40 candidates kept during selection
#define SCOL    (BB*NP*KNBR)  // 327680 flattened (b,n,k) columns
#define NKC     (NP*KNBR)     // 163840
#define CIN     10            // feature channels
#define CPAD    12            // padded to multiple of 4 for WMMA K-steps
#define CMID    64
#define EPS_PD  1e-6f
#define EPS_BN  1e-5f
#define SLOPE   0.2f

// ---------------- CDNA5 WMMA (wave32) ----------------
typedef __attribute__((ext_vector_type(2))) float v2f;
typedef __attribute__((ext_vector_type(8))) float v8f;

// D(16x16,f32) = A(16x4,f32) * B(4x16,f32) + C
// A layout: lanes 0-15 hold rows M=0..15; VGPR0/1 = K pair (half 0: K=0,1 ; half 1: K=2,3)
// C/D layout: VGPR r -> row M=r (lanes 0-15) / M=r+8 (lanes 16-31), col N = lane&15
__device__ __forceinline__ v8f wmma_f32_4(v2f a, v2f b, v8f c) {
    return __builtin_amdgcn_wmma_f32_16x16x4_f32(
        /*neg_a=*/false, a, /*neg_b=*/false, b,
        /*c_mod=*/(short)0, c, /*reuse_a=*/false, /*reuse_b=*/false);
}

// ---------------- helpers ----------------
__device__ __forceinline__ void atomicMaxF32(float* addr, float v) {
    // classic sign-split trick; valid with -inf initialization
    if (v >= 0.0f) atomicMax((int*)addr, __float_as_int(v));
    else           atomicMin((unsigned int*)addr, __float_as_uint(v));
}

__device__ __forceinline__ float halfwave_sum16(float v) {
    // sum across the 16 lanes of this half-wave (masks < 16 stay in group)
    v += __shfl_xor(v, 1);
    v += __shfl_xor(v, 2);
    v += __shfl_xor(v, 4);
    v += __shfl_xor(v, 8);
    return v;
}

__device__ __forceinline__ float lrelu(float v) {
    return (v >= 0.0f) ? v : SLOPE * v;
}

// ---------------- kernel 0: init output (-inf), stats (0), padded W1 ----------------
__global__ __launch_bounds__(256) void init_kernel(float* out, float* stats,
                                                   float* W1p, const float* __restrict__ W1,
                                                   int out_n) {
    int i = blockIdx.x * 256 + threadIdx.x;
    if (i < out_n) out[i] = -__builtin_huge_valf();
    if (i < 512)   stats[i] = 0.0f;
    if (i < CMID * CPAD) {                       // zero-padded W1: 64 x 12
        int row = i / CPAD, c = i - row * CPAD;
        W1p[i] = (c < CIN) ? W1[row * CIN + c] : 0.0f;
    }
}

// ---------------- kernel 1: brute-force dilated KNN ----------------
// one thread per query point; LDS-tiled candidate scan; top-40 kept sorted
// ascending in registers (stable: equal distances keep the earlier index).
__global__ __launch_bounds__(256) void knn_kernel(const float* __restrict__ pos,
                                                  int* __restrict__ idxb) {
    __shared__ float sx[1024], sy[1024], sz[1024];
    const int tid = threadIdx.x;
    const int b   = blockIdx.x >> 5;             // 32 blocks per batch
    const int i   = (blockIdx.x & 31) * 256 + tid;
    const float* px = pos + (b * 3 + 0) * NP;
    const float* py = pos + (b * 3 + 1) * NP;
    const float* pz = pos + (b * 3 + 2) * NP;
    const float xi = px[i], yi = py[i], zi = pz[i];

    float bd[CAND]; int bi[CAND];
#pragma unroll
    for (int t = 0; t < CAND; ++t) { bd[t] = 3.4e38f; bi[t] = 0; }

    for (int tile = 0; tile < NP / 1024; ++tile) {
        const int base = tile * 1024;
#pragma unroll
        for (int u = 0; u < 4; ++u) {
            int jj = tid + u * 256;
            sx[jj] = px[base + jj];
            sy[jj] = py[base + jj];
            sz[jj] = pz[base + jj];
        }
        __syncthreads();
        for (int jj = 0; jj < 1024; ++jj) {
            float dx = xi - sx[jj], dy = yi - sy[jj], dz = zi - sz[jj];
            float d2 = dx * dx + dy * dy + dz * dz;
            if (d2 < bd[CAND - 1]) {
                float cd = d2; int ci = base + jj;
#pragma unroll
                for (int t = 0; t < CAND; ++t) {
                    bool sw = cd < bd[t];
                    float td = bd[t]; int ti = bi[t];
                    bd[t] = sw ? cd : td;  bi[t] = sw ? ci : ti;
                    cd    = sw ? td : cd;  ci    = sw ? ti : ci;
                }
            }
        }
        __syncthreads();
    }
#pragma unroll
    for (int k = 0; k < KNBR; ++k)
        idxb[(b * NP + i) * KNBR + k] = bi[RATE * k];   // dilation ::2
}

// ---------------- kernel 2: build 12-row (padded) feature matrix ----------------
__global__ __launch_bounds__(256) void feat_kernel(const float* __restrict__ x,
                                                   const float* __restrict__ pos,
                                                   const int*   __restrict__ idxb,
                                                   float* __restrict__ feat) {
    int s = blockIdx.x * 256 + threadIdx.x;
    if (s >= SCOL) return;
    int b   = s / NKC;
    int rem = s - b * NKC;
    int n   = rem / KNBR;
    int j   = idxb[s];

    float xc0 = x[(b*3+0)*NP + n], xc1 = x[(b*3+1)*NP + n], xc2 = x[(b*3+2)*NP + n];
    float xn0 = x[(b*3+0)*NP + j], xn1 = x[(b*3+1)*NP + j], xn2 = x[(b*3+2)*NP + j];
    float pc0 = pos[(b*3+0)*NP + n], pc1 = pos[(b*3+1)*NP + n], pc2 = pos[(b*3+2)*NP + n];
    float pn0 = pos[(b*3+0)*NP + j], pn1 = pos[(b*3+1)*NP + j], pn2 = pos[(b*3+2)*NP + j];

    float dx = pc0 - pn0 + EPS_PD, dy = pc1 - pn1 + EPS_PD, dz = pc2 - pn2 + EPS_PD;
    float d  = sqrtf(dx * dx + dy * dy + dz * dz);

    feat[ 0*SCOL + s] = xn0 - xc0;
    feat[ 1*SCOL + s] = xn1 - xc1;
    feat[ 2*SCOL + s] = xn2 - xc2;
    feat[ 3*SCOL + s] = pn0 - pc0;
    feat[ 4*SCOL + s] = pn1 - pc1;
    feat[ 5*SCOL + s] = pn2 - pc2;
    feat[ 6*SCOL + s] = d;
    feat[ 7*SCOL + s] = xc0;
    feat[ 8*SCOL + s] = xc1;
    feat[ 9*SCOL + s] = xc2;
    feat[10*SCOL + s] = 0.0f;   // K padding for WMMA
    feat[11*SCOL + s] = 0.0f;
}

// ---------------- kernel 3: GEMM1 via f32 WMMA, store y1 + BN1 stats ----------------
// one wave per 16-wide s-tile; 4 accumulators cover all 64 output channels so
// every B load is reused by 4 WMMAs. A comes from the zero-padded W1 (64x12),
// so A-loads are unconditional b64 (rows of 12, even ci0 -> even flat index).
__global__ __launch_bounds__(256) void gemm1_kernel(const float* __restrict__ W1p,
                                                    const float* __restrict__ b1,
                                                    const float* __restrict__ feat,
                                                    float* __restrict__ y1,
                                                    float* __restrict__ sum,
                                                    float* __restrict__ sq) {
    int wave = (blockIdx.x * 256 + threadIdx.x) >> 5;
    int lane = threadIdx.x & 31;
    int half = lane >> 4, l = lane & 15;
    int s0   = wave * 16;
    const float2* W1v = (const float2*)W1p;

    v8f acc[4];
#pragma unroll
    for (int ot = 0; ot < 4; ++ot) acc[ot] = {};

#pragma unroll
    for (int kk = 0; kk < CPAD; kk += 4) {
        int ci0 = kk + (half ? 2 : 0);
        int ci1 = ci0 + 1;
        v2f bv;
        bv.x = feat[ci0 * SCOL + s0 + l];
        bv.y = feat[ci1 * SCOL + s0 + l];
#pragma unroll
        for (int ot = 0; ot < 4; ++ot) {
            int row = ot * 16 + l;
            float2 av = W1v[(row * CPAD + ci0) >> 1];
            v2f a; a.x = av.x; a.y = av.y;
            acc[ot] = wmma_f32_4(a, bv, acc[ot]);
        }
    }

#pragma unroll
    for (int ot = 0; ot < 4; ++ot) {
#pragma unroll
        for (int r = 0; r < 8; ++r) {
            int o = ot * 16 + r + 8 * half;
            float v = acc[ot][r] + b1[o];
            y1[o * SCOL + s0 + l] = v;
            float srd = halfwave_sum16(v);
            float qrd = halfwave_sum16(v * v);
            if (l == 0) { atomicAdd(&sum[o], srd); atomicAdd(&sq[o], qrd); }
        }
    }
}

// ---------------- kernel 4: BN finalize (scale/shift from sums) ----------------
__global__ __launch_bounds__(64) void bn_finalize_kernel(const float* __restrict__ sum,
                                                         const float* __restrict__ sq,
                                                         const float* __restrict__ g,
                                                         const float* __restrict__ be,
                                                         float* __restrict__ scale,
                                                         float* __restrict__ shift) {
    int o = threadIdx.x;
    const float inv = 1.0f / (float)SCOL;
    float m  = sum[o] * inv;
    float v  = sq[o] * inv - m * m;
    float sc = g[o] * rsqrtf(v + EPS_BN);
    scale[o] = sc;
    shift[o] = be[o] - m * sc;
}

// ---------------- kernels 5/7: GEMM2 via f32 WMMA ----------------
// BN1+LeakyReLU is fused into the B-operand load (no separate activation pass).
// One wave per s-tile with 4 accumulators (64 WMMAs per wave).
// MODE 0: accumulate BN2 stats only.  MODE 1: BN2+LReLU+max-over-K into out.
template <int MODE>
__global__ __launch_bounds__(256) void gemm2_kernel(const float* __restrict__ W2,
                                                    const float* __restrict__ b2,
                                                    const float* __restrict__ y1,
                                                    const float* __restrict__ scale1,
                                                    const float* __restrict__ shift1,
                                                    const float* __restrict__ scale2,
                                                    const float* __restrict__ shift2,
                                                    float* __restrict__ sum,
                                                    float* __restrict__ sq,
                                                    float* __restrict__ out) {
    int wave = (blockIdx.x * 256 + threadIdx.x) >> 5;
    int lane = threadIdx.x & 31;
    int half = lane >> 4, l = lane & 15;
    int s0   = wave * 16;
    const float2* W2v = (const float2*)W2;   // K-contiguous pair -> b64 load

    v8f acc[4];
#pragma unroll
    for (int ot = 0; ot < 4; ++ot) acc[ot] = {};

#pragma unroll
    for (int kk = 0; kk < CMID; kk += 4) {
        int ci0 = kk + (half ? 2 : 0);
        int ci1 = ci0 + 1;
        // B operand: h1 = lrelu(bn1(y1)) computed on the fly
        float r0 = y1[ci0 * SCOL + s0 + l];
        float r1 = y1[ci1 * SCOL + s0 + l];
        r0 = lrelu(r0 * scale1[ci0] + shift1[ci0]);
        r1 = lrelu(r1 * scale1[ci1] + shift1[ci1]);
        v2f bv; bv.x = r0; bv.y = r1;
#pragma unroll
        for (int ot = 0; ot < 4; ++ot) {
            int row = ot * 16 + l;
            float2 av = W2v[(row * CMID + ci0) >> 1];
            v2f a; a.x = av.x; a.y = av.y;
            acc[ot] = wmma_f32_4(a, bv, acc[ot]);
        }
    }

    int scol = s0 + l;
    int nf   = scol / KNBR;          // flattened (b,n)
    int bidx = nf >> 13;             // / 8192
    int n    = nf & (NP - 1);

#pragma unroll
    for (int ot = 0; ot < 4; ++ot) {
#pragma unroll
        for (int r = 0; r < 8; ++r) {
            int o = ot * 16 + r + 8 * half;
            float v = acc[ot][r] + b2[o];
            if (MODE == 0) {
                float srd = halfwave_sum16(v);
                float qrd = halfwave_sum16(v * v);
                if (l == 0) { atomicAdd(&sum[o], srd); atomicAdd(&sq[o], qrd); }
            } else {
                v = lrelu(v * scale2[o] + shift2[o]);
                atomicMaxF32(&out[(bidx * CMID + o) * NP + n], v);
            }
        }
    }
}

// ---------------- host launcher ----------------
extern "C" void kernel_launch(void* const* d_in, const int* in_sizes, int n_in,
                              void* d_out, int out_size, void* d_ws, size_t ws_size,
                              hipStream_t stream) {
    const float* x   = (const float*)d_in[0];
    const float* pos = (const float*)d_in[1];
    const float* W1  = (const float*)d_in[2];
    const float* b1  = (const float*)d_in[3];
    const float* g1  = (const float*)d_in[4];
    const float* be1 = (const float*)d_in[5];
    const float* W2  = (const float*)d_in[6];
    const float* b2  = (const float*)d_in[7];
    const float* g2  = (const float*)d_in[8];
    const float* be2 = (const float*)d_in[9];
    float* out = (float*)d_out;

    // workspace layout (bytes):
    //   [0, 2048)            : stats -> sum1|sq1|sum2|sq2|scale1|shift1|scale2|shift2 (8x64 f32)
    //   [2048, 5120)         : W1p   (64 x 12 f32, zero padded)
    //   [5120, +1310720)     : idx   (SCOL int32)
    //   [.., +15728640)      : feat  (12 x SCOL f32)
    //   [.., +83886080)      : y1    (64 x SCOL f32, raw pre-BN)
    char* ws = (char*)d_ws;
    float* stats  = (float*)ws;
    float* sum1   = stats +   0;
    float* sq1    = stats +  64;
    float* sum2   = stats + 128;
    float* sq2    = stats + 192;
    float* scale1 = stats + 256;
    float* shift1 = stats + 320;
    float* scale2 = stats + 384;
    float* shift2 = stats + 448;
    float* W1p    = (float*)(ws + 2048);
    int*   idxb   = (int*)  (ws + 5120);
    float* feat   = (float*)(ws + 5120 + (size_t)SCOL * 4);
    float* y1     = (float*)(ws + 5120 + (size_t)SCOL * 4 + (size_t)CPAD * SCOL * 4);

    const int gemm_blocks = (SCOL / 16) / 8;   // 20480 s-tile waves / 8 waves per block

    init_kernel<<<(out_size + 255) / 256, 256, 0, stream>>>(out, stats, W1p, W1, out_size);
    knn_kernel<<<(BB * NP) / 256, 256, 0, stream>>>(pos, idxb);
    feat_kernel<<<SCOL / 256, 256, 0, stream>>>(x, pos, idxb, feat);
    gemm1_kernel<<<gemm_blocks, 256, 0, stream>>>(W1p, b1, feat, y1, sum1, sq1);
    bn_finalize_kernel<<<1, 64, 0, stream>>>(sum1, sq1, g1, be1, scale1, shift1);
    gemm2_kernel<0><<<gemm_blocks, 256, 0, stream>>>(W2, b2, y1, scale1, shift1,
                                                     scale2, shift2, sum2, sq2, out);
    bn_finalize_kernel<<<1, 64, 0, stream>>>(sum2, sq2, g2, be2, scale2, shift2);
    gemm2_kernel<1><<<gemm_blocks, 256, 0, stream>>>(W2, b2, y1, scale1, shift1,
                                                     scale2, shift2, sum2, sq2, out);
}